// QuantizedWeight_52106543235656
// MI455X (gfx1250) — compile-verified
//
#include <hip/hip_runtime.h>
#include <hip/hip_bf16.h>
#include <stdint.h>

// AQLM dequant: out[o, i*8+j] = scales[o] * (cb0[codes[o,i,0]][j] + cb1[codes[o,i,1]][j])
// O=8192 rows, I=1024 i-groups, 2 codebooks of 65536 x 8 fp32 entries (32 B each).
//
// Roofline: ~324 MB HBM traffic, 0.13 GFLOP -> pure bandwidth/gather kernel (~14 us
// floor at 23.3 TB/s). Codebooks (4 MB) are L2-resident; all streaming traffic
// (codes in, output out) uses non-temporal hints so it can't evict them.
//
//  - codes (8 KB/row) staged to LDS via CDNA5 async global->LDS, th:TH_LOAD_NT
//  - codebook gathers use uniform-base + 16-bit-masked u32 offsets -> SADDR form
//  - output written with non-temporal b128 stores

typedef float v4f __attribute__((ext_vector_type(4)));

#define O_GROUPS 8192
#define I_GROUPS 1024
#define N_CB     2
#define CB_SIZE  65536
#define IG       8
#define THREADS  256
#define IPT      (I_GROUPS / THREADS)   // 4 i-groups per thread

__global__ __launch_bounds__(THREADS)
void aqlm_dequant_kernel(const int*   __restrict__ codes,      // [O, I, 2] int32
                         const float* __restrict__ codebooks,  // [2, 65536, 1, 8] f32
                         const float* __restrict__ scales,     // [O] f32
                         float*       __restrict__ out)        // [O, I*8] f32
{
    __shared__ int lds_codes[I_GROUPS * N_CB];   // 8 KB per block

    const int o   = blockIdx.x;     // one output row per block (uniform)
    const int tid = threadIdx.x;

    // ---- Stage this row's codes (8 KB contiguous) into LDS asynchronously ----
    // Two block-wide GLOBAL_LOAD_ASYNC_TO_LDS_B128 issues: 256 lanes x 16 B = 4 KB each.
    // Non-temporal: the 64 MB code stream is read-once.
    {
        const uint64_t gbase = (uint64_t)(uintptr_t)(codes + (size_t)o * (I_GROUPS * N_CB));
        const uint32_t goff  = (uint32_t)tid * 16u;                       // byte offset
        const uint32_t loff  = (uint32_t)(uintptr_t)lds_codes + goff;     // LDS byte addr
        asm volatile("global_load_async_to_lds_b128 %0, %1, %2 th:TH_LOAD_NT"
                     :: "v"(loff), "v"(goff), "s"(gbase) : "memory");
        asm volatile("global_load_async_to_lds_b128 %0, %1, %2 th:TH_LOAD_NT"
                     :: "v"(loff + 4096u), "v"(goff + 4096u), "s"(gbase) : "memory");
        asm volatile("s_wait_asynccnt 0" ::: "memory");
    }
    __syncthreads();

    const float s = scales[o];                          // block-uniform -> s_load
    const char* cb0 = (const char*)codebooks;
    const char* cb1 = (const char*)(codebooks + (size_t)CB_SIZE * IG);
    char* orow = (char*)(out + (size_t)o * (I_GROUPS * IG));

#pragma unroll
    for (int k = 0; k < IPT; ++k) {
        const int i = tid + k * THREADS;
        // mask to 16 bits: offsets provably < 2^21 -> zext==sext -> SADDR gather form
        const uint32_t c0 = ((uint32_t)lds_codes[2 * i + 0]) & 0xFFFFu;
        const uint32_t c1 = ((uint32_t)lds_codes[2 * i + 1]) & 0xFFFFu;
        const uint32_t b0 = c0 << 5;        // byte offset into codebook 0 (<= 2 MB)
        const uint32_t b1 = c1 << 5;

        // 16-B gathers from the L2-resident codebooks (global_load_b128, saddr+voffset)
        v4f e0lo = *(const v4f*)(cb0 + b0);
        v4f e0hi = *(const v4f*)(cb0 + b0 + 16);
        v4f e1lo = *(const v4f*)(cb1 + b1);
        v4f e1hi = *(const v4f*)(cb1 + b1 + 16);

        v4f lo = (e0lo + e1lo) * s;
        v4f hi = (e0hi + e1hi) * s;

        // Streaming result: non-temporal b128 stores (keep codebooks hot in L2)
        const uint32_t ob = (uint32_t)i << 5;            // < 32 KB row offset
        __builtin_nontemporal_store(lo, (v4f*)(orow + ob));
        __builtin_nontemporal_store(hi, (v4f*)(orow + ob + 16));
    }
}

extern "C" void kernel_launch(void* const* d_in, const int* in_sizes, int n_in,
                              void* d_out, int out_size, void* d_ws, size_t ws_size,
                              hipStream_t stream) {
    (void)in_sizes; (void)n_in; (void)d_ws; (void)ws_size; (void)out_size;
    const int*   codes     = (const int*)  d_in[0];   // [8192, 1024, 2]
    const float* codebooks = (const float*)d_in[1];   // [2, 65536, 1, 8]
    const float* scales    = (const float*)d_in[2];   // [8192, 1, 1, 1]
    float*       out       = (float*)d_out;           // [8192, 8192]

    aqlm_dequant_kernel<<<dim3(O_GROUPS), dim3(THREADS), 0, stream>>>(
        codes, codebooks, scales, out);
}